// MySCNN_30691836297642
// MI455X (gfx1250) — compile-verified
//
#include <hip/hip_runtime.h>
#include <math.h>

#define M0   200000
#define BB   4
#define KCH  5
#define FF   10
#define SLOPE 0.01f

typedef __attribute__((ext_vector_type(2))) float v2f;
typedef __attribute__((ext_vector_type(8))) float v8f;

// ---------------- elementwise helpers ----------------

__global__ void zero_kernel(float* __restrict__ p, size_t n) {
  size_t i = (size_t)blockIdx.x * blockDim.x + threadIdx.x;
  if (i < n) p[i] = 0.f;
}

// z = 2*z - xprev   (Chebyshev recurrence combine, in place over the spmv result)
__global__ void combine_kernel(float* __restrict__ z, const float* __restrict__ xprev, size_t n) {
  size_t i = (size_t)blockIdx.x * blockDim.x + threadIdx.x;
  if (i < n) z[i] = 2.f * z[i] - xprev[i];
}

// ---------------- SPMV: COO scatter, channels innermost ----------------
// x, y layout: [(b*M0 + m)*C + i].  y must be pre-zeroed.
template<int C>
__global__ void spmv_scatter(const int* __restrict__ rows, const int* __restrict__ cols,
                             const float* __restrict__ vals, int E,
                             const float* __restrict__ x, float* __restrict__ y) {
  int e = blockIdx.x * blockDim.x + threadIdx.x;
  if (e >= E) return;
  int r = rows[e];
  int c = cols[e];
  float v = vals[e];
#pragma unroll
  for (int b = 0; b < BB; ++b) {
    const float* xb = x + ((size_t)b * M0 + (size_t)c) * C;
    float*       yb = y + ((size_t)b * M0 + (size_t)r) * C;
#pragma unroll
    for (int i = 0; i < C; ++i) {
      unsafeAtomicAdd(yb + i, v * xb[i]);
    }
  }
}

// ---------------- Chebyshev channel-mix GEMM via f32 WMMA ----------------
// out[r*CO + o] = act( sum_{k,i} term_k[r*CI + i] * theta[(o*CI+i)*K + k] + bias[o] )
// One wave computes a 16(row) x 16(col) tile with v_wmma_f32_16x16x4_f32.
// theta is staged ZERO-PADDED in LDS; the zero B rows/cols null out any padded
// A lanes, so every A load is unconditional (no selects, no EXEC divergence).
template<int CI, int CO, int ACT>
__global__ void __launch_bounds__(256)
cheb_mix_wmma(const float* __restrict__ t0, const float* __restrict__ t1,
              const float* __restrict__ t2, const float* __restrict__ t3,
              const float* __restrict__ t4,
              const float* __restrict__ theta, const float* __restrict__ bias,
              float* __restrict__ out, int R) {
  constexpr int CIP = (CI + 3) & ~3;               // contraction padded to mult of 4
  __shared__ float Bs[KCH * CIP * 16];             // [k][j][n], zero padded

  // cooperative zero-padded theta stage: Bs[(k*CIP + j)*16 + n]
  for (int idx = threadIdx.x; idx < KCH * CIP * 16; idx += blockDim.x) {
    int n = idx & 15;
    int t = idx >> 4;
    int j = t % CIP;
    int k = t / CIP;
    Bs[idx] = (n < CO && j < CI) ? theta[((size_t)n * CI + (size_t)j) * KCH + k] : 0.f;
  }
  __syncthreads();

  const float* terms[KCH] = { t0, t1, t2, t3, t4 };

  int lane = threadIdx.x & 31;
  int wave = threadIdx.x >> 5;
  int tile = blockIdx.x * (blockDim.x >> 5) + wave;
  int row0 = tile * 16;
  if (row0 >= R) return;          // wave-uniform (grid is exact); EXEC stays all-ones

  int la = lane & 15;             // N column (output channel) / A row within half
  int hi = lane >> 4;             // 0: k-pair {kk,kk+1}, 1: {kk+2,kk+3}
  size_t arow = (size_t)(row0 + la) * CI;

  v8f acc = {0.f, 0.f, 0.f, 0.f, 0.f, 0.f, 0.f, 0.f};

#pragma unroll
  for (int k = 0; k < KCH; ++k) {
    const float* __restrict__ A = terms[k];
#pragma unroll
    for (int kk = 0; kk < CI; kk += 4) {
      int j0 = kk + hi * 2;
      v2f a;
      if constexpr (CI == 1) {
        // Every lane loads the (always valid) single element; contributions of
        // j=1..3 are nulled by the zero rows of Bs.
        float av = A[arow];
        a.x = av;
        a.y = av;
      } else {
        // CI even; clamp to an always-valid even pair. Padded (j>=CI) lanes are
        // nulled by the zero rows of Bs, so no value masking is needed.
        int j0c = (j0 + 1 < CI) ? j0 : (CI - 2);
        a = *(const v2f*)(A + arow + (size_t)j0c);        // 8B-aligned, unconditional
      }
      v2f bm;
      bm.x = Bs[(k * CIP + j0) * 16 + la];
      bm.y = Bs[(k * CIP + j0 + 1) * 16 + la];
      acc = __builtin_amdgcn_wmma_f32_16x16x4_f32(false, a, false, bm, (short)0, acc,
                                                  false, false);
    }
  }

  float bo = bias[(la < CO) ? la : (CO - 1)];      // clamped, unconditional load
#pragma unroll
  for (int v = 0; v < 8; ++v) {
    int r = row0 + v + hi * 8;    // D layout: VGPR v holds M = v (+8 for lanes 16..31)
    if (la < CO) {
      float val = acc[v] + bo;
      if (ACT) val = (val > 0.f) ? val : SLOPE * val;
      out[(size_t)r * CO + la] = val;
    }
  }
}

// ---------------- final 3->2 FC + sigmoid ----------------
__global__ void final_fc(const float* __restrict__ o0, const float* __restrict__ o1,
                         const float* __restrict__ o2,
                         const float* __restrict__ fcw, const float* __restrict__ fcb,
                         float* __restrict__ out, int R) {
  int r = blockIdx.x * blockDim.x + threadIdx.x;
  if (r >= R) return;
  float c0 = o0[r], c1 = o1[r], c2 = o2[r];
#pragma unroll
  for (int o = 0; o < 2; ++o) {
    float l = fcw[o * 3 + 0] * c0 + fcw[o * 3 + 1] * c1 + fcw[o * 3 + 2] * c2 + fcb[o];
    out[(size_t)r * 2 + o] = 1.f / (1.f + expf(-l));
  }
}

// ---------------- host-side layer driver ----------------
template<int CI, int CO, int ACT>
static void run_layer(const int* rows, const int* cols, const float* vals, int E,
                      const float* xin, float* T0, float* T1, float* T2, float* T3,
                      const float* th, const float* bi, float* out, hipStream_t s) {
  const size_t N = (size_t)BB * M0 * CI;
  const int R = BB * M0;
  const int zb = (int)((N + 255) / 256);
  const int sb = (E + 255) / 256;

  // X1 = L x0
  zero_kernel<<<zb, 256, 0, s>>>(T0, N);
  spmv_scatter<CI><<<sb, 256, 0, s>>>(rows, cols, vals, E, xin, T0);
  // X2 = 2 L X1 - X0
  zero_kernel<<<zb, 256, 0, s>>>(T1, N);
  spmv_scatter<CI><<<sb, 256, 0, s>>>(rows, cols, vals, E, T0, T1);
  combine_kernel<<<zb, 256, 0, s>>>(T1, xin, N);
  // X3 = 2 L X2 - X1
  zero_kernel<<<zb, 256, 0, s>>>(T2, N);
  spmv_scatter<CI><<<sb, 256, 0, s>>>(rows, cols, vals, E, T1, T2);
  combine_kernel<<<zb, 256, 0, s>>>(T2, T0, N);
  // X4 = 2 L X3 - X2
  zero_kernel<<<zb, 256, 0, s>>>(T3, N);
  spmv_scatter<CI><<<sb, 256, 0, s>>>(rows, cols, vals, E, T2, T3);
  combine_kernel<<<zb, 256, 0, s>>>(T3, T1, N);
  // mix all K terms with WMMA, fused bias + activation
  cheb_mix_wmma<CI, CO, ACT><<<(R + 127) / 128, 256, 0, s>>>(xin, T0, T1, T2, T3,
                                                             th, bi, out, R);
}

extern "C" void kernel_launch(void* const* d_in, const int* in_sizes, int n_in,
                              void* d_out, int out_size, void* d_ws, size_t ws_size,
                              hipStream_t stream) {
  (void)n_in; (void)out_size; (void)ws_size;

  const float* x[3]    = { (const float*)d_in[0], (const float*)d_in[1], (const float*)d_in[2] };
  const int*   rows[3] = { (const int*)d_in[3], (const int*)d_in[6], (const int*)d_in[9] };
  const int*   cols[3] = { (const int*)d_in[4], (const int*)d_in[7], (const int*)d_in[10] };
  const float* vals[3] = { (const float*)d_in[5], (const float*)d_in[8], (const float*)d_in[11] };
  int E[3] = { in_sizes[3], in_sizes[6], in_sizes[9] };

  const float *th1[3], *b1[3], *th2[3], *b2[3], *th3[3], *b3[3];
  for (int d = 0; d < 3; ++d) {
    int base = 12 + d * 6;
    th1[d] = (const float*)d_in[base + 0];
    b1[d]  = (const float*)d_in[base + 1];
    th2[d] = (const float*)d_in[base + 2];
    b2[d]  = (const float*)d_in[base + 3];
    th3[d] = (const float*)d_in[base + 4];
    b3[d]  = (const float*)d_in[base + 5];
  }
  const float* fcw = (const float*)d_in[30];
  const float* fcb = (const float*)d_in[31];

  // workspace carve-up (floats)
  float* ws = (float*)d_ws;
  const size_t TERM = (size_t)BB * M0 * FF;     // 8M floats / 32 MB
  float* T0  = ws;
  float* T1  = ws + TERM;
  float* T2  = ws + 2 * TERM;
  float* T3  = ws + 3 * TERM;
  float* LOA = ws + 4 * TERM;                   // layer-1 output / layer-2 input
  float* LOB = ws + 5 * TERM;                   // layer-2 output / layer-3 input
  float* BO  = ws + 6 * TERM;                   // 3 x (B*M) branch outputs

  const int R = BB * M0;
  for (int d = 0; d < 3; ++d) {
    float* bout = BO + (size_t)d * R;
    run_layer<1,  FF, 1>(rows[d], cols[d], vals[d], E[d], x[d], T0, T1, T2, T3,
                         th1[d], b1[d], LOA, stream);
    run_layer<FF, FF, 1>(rows[d], cols[d], vals[d], E[d], LOA,  T0, T1, T2, T3,
                         th2[d], b2[d], LOB, stream);
    run_layer<FF, 1,  0>(rows[d], cols[d], vals[d], E[d], LOB,  T0, T1, T2, T3,
                         th3[d], b3[d], bout, stream);
  }

  final_fc<<<(R + 255) / 256, 256, 0, stream>>>(BO, BO + R, BO + 2 * (size_t)R,
                                                fcw, fcb, (float*)d_out, R);
}